// TemporalBlock_5669356831787
// MI455X (gfx1250) — compile-verified
//
#include <hip/hip_runtime.h>
#include <math.h>

// Problem constants (match reference)
#define B_    16
#define C_    256
#define T_    1024
#define KEY_  256
#define KS_   3
#define PAD_  2
// sqrt(KEY) == 16 exactly
#define INV_SQRT_K (1.0f / 16.0f)

typedef __attribute__((ext_vector_type(2))) float v2f;
typedef __attribute__((ext_vector_type(8))) float v8f;

// D = A(16x4 f32) * B(4x16 f32) + C(16x16 f32), wave32.
// A frag: lane<16 -> {A[m][k], A[m][k+1]}, lane>=16 -> {A[m][k+2], A[m][k+3]}, m=lane&15
// B frag: lane<16 -> {B[k][n], B[k+1][n]},  lane>=16 -> {B[k+2][n], B[k+3][n]}, n=lane&15
// C/D:    vgpr r -> (row = r + (lane<16?0:8), col = lane&15)
__device__ __forceinline__ v8f wmma4(v2f a, v2f b, v8f c) {
  return __builtin_amdgcn_wmma_f32_16x16x4_f32(
      /*neg_a=*/false, a, /*neg_b=*/false, b,
      /*c_mod=*/(short)0, c, /*reuse_a=*/false, /*reuse_b=*/false);
}

// ---------------------------------------------------------------------------
// K0: weight_norm (dim=0) + transpose conv weights into wt[kk][i][o]
// so conv-GEMM A-fragments are lane-coalesced.
// ---------------------------------------------------------------------------
__global__ void wnorm_kernel(const float* __restrict__ v,
                             const float* __restrict__ g,
                             float* __restrict__ wt) {
  int o = blockIdx.x * blockDim.x + threadIdx.x;
  if (o >= C_) return;
  const float* vo = v + (size_t)o * C_ * KS_;
  float ss = 0.f;
  for (int i = 0; i < C_ * KS_; ++i) { float w = vo[i]; ss += w * w; }
  float scale = g[o] / sqrtf(ss);
  for (int i = 0; i < C_; ++i)
    for (int kk = 0; kk < KS_; ++kk)
      wt[(size_t)kk * C_ * C_ + (size_t)i * C_ + o] = scale * vo[i * KS_ + kk];
}

// ---------------------------------------------------------------------------
// K1: q/k/v projections, 16(t) x 64(n) tile per wave: the A fragment
// (streamed x) is shared across 4 N-subtiles -> 4x reuse of the HBM stream.
// ---------------------------------------------------------------------------
__global__ void qkv_kernel(const float* __restrict__ x,
                           const float* __restrict__ Wq, const float* __restrict__ bq,
                           const float* __restrict__ Wk, const float* __restrict__ bk,
                           const float* __restrict__ Wv, const float* __restrict__ bv,
                           float* __restrict__ q, float* __restrict__ k,
                           float* __restrict__ vout) {
  const int lane = threadIdx.x;
  const int t0 = (blockIdx.x >> 2) << 4;   // 64 t-tiles
  const int n0 = (blockIdx.x & 3) << 6;    // 4 n-macrotiles of 64
  const int b  = blockIdx.z;
  const float* W; const float* bias; float* out;
  if (blockIdx.y == 0)      { W = Wq; bias = bq; out = q;    }
  else if (blockIdx.y == 1) { W = Wk; bias = bk; out = k;    }
  else                      { W = Wv; bias = bv; out = vout; }
  const float* xb = x + (size_t)b * C_ * T_;
  float* ob = out + (size_t)b * T_ * 256;
  const int m = lane & 15;
  const int koff = (lane < 16) ? 0 : 2;
  v8f acc[4] = {};
  for (int kk = 0; kk < C_; kk += 4) {
    const int kb = kk + koff;
    v2f a;
    a.x = xb[(size_t)kb * T_ + t0 + m];
    a.y = xb[(size_t)(kb + 1) * T_ + t0 + m];
#pragma unroll
    for (int j = 0; j < 4; ++j) {
      v2f bb;
      bb.x = W[(size_t)kb * 256 + n0 + j * 16 + m];
      bb.y = W[(size_t)(kb + 1) * 256 + n0 + j * 16 + m];
      acc[j] = wmma4(a, bb, acc[j]);
    }
  }
  const int rofs = (lane < 16) ? 0 : 8;
#pragma unroll
  for (int j = 0; j < 4; ++j)
    for (int r = 0; r < 8; ++r) {
      const int row = t0 + r + rofs;
      const int col = n0 + j * 16 + m;
      ob[(size_t)row * 256 + col] = acc[j][r] + bias[col];
    }
}

// ---------------------------------------------------------------------------
// K2: scores stored KEY-major: SC[b][s][t] = (q[t].k[s])/16, -inf where s>t.
// 64(s) x 16(t) tile per wave; k/q tiles staged in LDS (20 KB), the shared
// B fragment (q) feeds 4 accumulators. Fully-masked macrotiles skip the GEMM.
// ---------------------------------------------------------------------------
__global__ void scores_kernel(const float* __restrict__ q,
                              const float* __restrict__ k,
                              float* __restrict__ sc) {
  __shared__ float lk[64 * 64];
  __shared__ float lq[16 * 64];
  const int lane = threadIdx.x;
  const int st = blockIdx.x >> 6;          // 16 s-macrotiles
  const int tt = blockIdx.x & 63;          // 64 t-tiles
  const int b  = blockIdx.z;
  const int s0 = st << 6, t0 = tt << 4;
  const int m = lane & 15;
  const int rofs = (lane < 16) ? 0 : 8;
  float* scb = sc + (size_t)b * T_ * T_;
  if (s0 > t0 + 15) {  // whole macrotile masked
#pragma unroll
    for (int j = 0; j < 4; ++j)
      for (int r = 0; r < 8; ++r)
        scb[(size_t)(s0 + j * 16 + r + rofs) * T_ + t0 + m] = -INFINITY;
    return;
  }
  const float* qb = q + (size_t)b * T_ * KEY_;
  const float* kb_ = k + (size_t)b * T_ * KEY_;
  const int koff = (lane < 16) ? 0 : 2;
  v8f acc[4] = {};
  for (int c0 = 0; c0 < KEY_; c0 += 64) {
    for (int it = 0; it < 32; ++it) {      // 1024 float4 for the 64x64 k tile
      const int idx = lane + it * 32;
      const int row = idx >> 4, c4 = idx & 15;
      *(float4*)&lk[row * 64 + c4 * 4] =
          *(const float4*)&kb_[(size_t)(s0 + row) * KEY_ + c0 + c4 * 4];
    }
    for (int it = 0; it < 8; ++it) {       // 256 float4 for the 16x64 q tile
      const int idx = lane + it * 32;
      const int row = idx >> 4, c4 = idx & 15;
      *(float4*)&lq[row * 64 + c4 * 4] =
          *(const float4*)&qb[(size_t)(t0 + row) * KEY_ + c0 + c4 * 4];
    }
    __syncthreads();
    for (int kk = 0; kk < 64; kk += 4) {
      const int kb = kk + koff;
      v2f bb;
      bb.x = lq[m * 64 + kb];
      bb.y = lq[m * 64 + kb + 1];
#pragma unroll
      for (int j = 0; j < 4; ++j) {
        v2f a;
        a.x = lk[(j * 16 + m) * 64 + kb];
        a.y = lk[(j * 16 + m) * 64 + kb + 1];
        acc[j] = wmma4(a, bb, acc[j]);
      }
    }
    __syncthreads();
  }
#pragma unroll
  for (int j = 0; j < 4; ++j)
    for (int r = 0; r < 8; ++r) {
      const int s = s0 + j * 16 + r + rofs;
      const int t = t0 + m;
      scb[(size_t)s * T_ + t] = (s > t) ? -INFINITY : acc[j][r] * INV_SQRT_K;
    }
}

// ---------------------------------------------------------------------------
// K3: softmax over the QUERY axis t (reference softmaxes axis=1).
// SC is [s][t], so this is a contiguous row softmax. In-place -> AW.
// ---------------------------------------------------------------------------
__global__ void softmax_col_kernel(float* __restrict__ sc) {
  __shared__ float red[256];
  const int tid = threadIdx.x;
  float* row = sc + (size_t)blockIdx.y * T_ * T_ + (size_t)blockIdx.x * T_;
  float mx = -INFINITY;
  for (int t = tid; t < T_; t += 256) mx = fmaxf(mx, row[t]);
  red[tid] = mx; __syncthreads();
  for (int off = 128; off > 0; off >>= 1) {
    if (tid < off) red[tid] = fmaxf(red[tid], red[tid + off]);
    __syncthreads();
  }
  mx = red[0]; __syncthreads();
  float sum = 0.f;
  for (int t = tid; t < T_; t += 256) {
    const float e = __expf(row[t] - mx);
    row[t] = e;
    sum += e;
  }
  red[tid] = sum; __syncthreads();
  for (int off = 128; off > 0; off >>= 1) {
    if (tid < off) red[tid] += red[tid + off];
    __syncthreads();
  }
  const float inv = 1.0f / red[0];
  for (int t = tid; t < T_; t += 256) row[t] *= inv;
}

// K4: rowsum over keys i of attn_w[t][i] == column-sum over s of AW[s][t]
__global__ void attnsum_kernel(const float* __restrict__ aw, float* __restrict__ rs) {
  const int t = blockIdx.x * 256 + threadIdx.x;
  const float* a = aw + (size_t)blockIdx.y * T_ * T_;
  float s = 0.f;
  for (int i = 0; i < T_; ++i) s += a[(size_t)i * T_ + t];
  rs[blockIdx.y * T_ + t] = s;
}

// K5: weight_x = softmax over t of rowsums (per batch)
__global__ void wx_kernel(const float* __restrict__ rs, float* __restrict__ wx) {
  __shared__ float red[256];
  const int tid = threadIdx.x;
  const float* r = rs + (size_t)blockIdx.x * T_;
  float* w = wx + (size_t)blockIdx.x * T_;
  float mx = -INFINITY;
  for (int t = tid; t < T_; t += 256) mx = fmaxf(mx, r[t]);
  red[tid] = mx; __syncthreads();
  for (int off = 128; off > 0; off >>= 1) {
    if (tid < off) red[tid] = fmaxf(red[tid], red[tid + off]);
    __syncthreads();
  }
  mx = red[0]; __syncthreads();
  float sum = 0.f;
  for (int t = tid; t < T_; t += 256) {
    const float e = __expf(r[t] - mx);
    w[t] = e;
    sum += e;
  }
  red[tid] = sum; __syncthreads();
  for (int off = 128; off > 0; off >>= 1) {
    if (tid < off) red[tid] += red[tid + off];
    __syncthreads();
  }
  const float inv = 1.0f / red[0];
  for (int t = tid; t < T_; t += 256) w[t] *= inv;
}

// ---------------------------------------------------------------------------
// K6: out_attn[b][c][j] = sum_i AW[i][j] * v[i][c] -- computed as O'[c][j].
// 64(c) x 16(j) tile per wave: the shared B fragment is the expensive AW
// stream (64 MB) -> AW traffic / 4. Contraction truncates at i <= j0+15
// (rows above are exactly zero after the causal softmax).
// ---------------------------------------------------------------------------
__global__ void attnv_kernel(const float* __restrict__ aw,
                             const float* __restrict__ v,
                             float* __restrict__ oa) {
  const int lane = threadIdx.x;
  const int c0 = (blockIdx.x >> 6) << 6;   // 4 c-macrotiles
  const int j0 = (blockIdx.x & 63) << 4;   // 64 j-tiles
  const int b = blockIdx.z;
  const float* ab = aw + (size_t)b * T_ * T_;
  const float* vb = v + (size_t)b * T_ * C_;
  const int m = lane & 15;
  const int koff = (lane < 16) ? 0 : 2;
  v8f acc[4] = {};
  const int imax = j0 + 16;  // AW[i][j]==0 for i>j
  for (int i = 0; i < imax; i += 4) {
    const int kb = i + koff;
    v2f bb;
    bb.x = ab[(size_t)kb * T_ + j0 + m];
    bb.y = ab[(size_t)(kb + 1) * T_ + j0 + m];
#pragma unroll
    for (int j = 0; j < 4; ++j) {
      v2f a;
      a.x = vb[(size_t)kb * C_ + c0 + j * 16 + m];
      a.y = vb[(size_t)(kb + 1) * C_ + c0 + j * 16 + m];
      acc[j] = wmma4(a, bb, acc[j]);
    }
  }
  float* ob = oa + (size_t)b * C_ * T_;
  const int rofs = (lane < 16) ? 0 : 8;
#pragma unroll
  for (int j = 0; j < 4; ++j)
    for (int r = 0; r < 8; ++r)
      ob[(size_t)(c0 + j * 16 + r + rofs) * T_ + j0 + m] = acc[j][r];
}

// ---------------------------------------------------------------------------
// K7/K8: causal conv (left-pad 2, K=3) as 3 shifted GEMMs over channels.
// 64(o) x 16(t) tile per wave: the shared B fragment is the activation
// stream h -> activation traffic / 4. final_stage=1 fuses the residual
// epilogue relu(relu(conv+b) + x + weight_x*x).
// ---------------------------------------------------------------------------
__global__ void conv_kernel(const float* __restrict__ h,
                            const float* __restrict__ wt,
                            const float* __restrict__ bias,
                            float* __restrict__ y,
                            const float* __restrict__ x,
                            const float* __restrict__ wx,
                            float* __restrict__ out,
                            int final_stage) {
  const int lane = threadIdx.x;
  const int o0 = (blockIdx.x >> 6) << 6;   // 4 o-macrotiles
  const int t0 = (blockIdx.x & 63) << 4;   // 64 t-tiles
  const int b = blockIdx.z;
  const float* hb = h + (size_t)b * C_ * T_;
  const int m = lane & 15;
  const int koff = (lane < 16) ? 0 : 2;
  v8f acc[4] = {};
  for (int kk = 0; kk < KS_; ++kk) {
    const float* wtk = wt + (size_t)kk * C_ * C_;
    const int ts = t0 + m - PAD_ + kk;  // shifted time index for this lane's col
    const bool ok = (ts >= 0);
    for (int i = 0; i < C_; i += 4) {
      const int kb = i + koff;
      v2f bb;
      bb.x = ok ? hb[(size_t)kb * T_ + ts] : 0.f;
      bb.y = ok ? hb[(size_t)(kb + 1) * T_ + ts] : 0.f;
#pragma unroll
      for (int j = 0; j < 4; ++j) {
        v2f a;
        a.x = wtk[(size_t)kb * C_ + o0 + j * 16 + m];
        a.y = wtk[(size_t)(kb + 1) * C_ + o0 + j * 16 + m];
        acc[j] = wmma4(a, bb, acc[j]);
      }
    }
  }
  const int rofs = (lane < 16) ? 0 : 8;
#pragma unroll
  for (int j = 0; j < 4; ++j)
    for (int r = 0; r < 8; ++r) {
      const int o = o0 + j * 16 + r + rofs;
      const int t = t0 + m;
      float vv = fmaxf(acc[j][r] + bias[o], 0.f);  // relu(conv + b)
      const size_t idx = (size_t)b * C_ * T_ + (size_t)o * T_ + t;
      if (final_stage) {
        const float xv = x[idx];
        out[idx] = fmaxf(vv + xv + wx[b * T_ + t] * xv, 0.f);
      } else {
        y[idx] = vv;
      }
    }
}

// ---------------------------------------------------------------------------
extern "C" void kernel_launch(void* const* d_in, const int* in_sizes, int n_in,
                              void* d_out, int out_size, void* d_ws, size_t ws_size,
                              hipStream_t stream) {
  const float* x       = (const float*)d_in[0];
  const float* Wq      = (const float*)d_in[1];
  const float* bq      = (const float*)d_in[2];
  const float* Wk      = (const float*)d_in[3];
  const float* bk      = (const float*)d_in[4];
  const float* Wv      = (const float*)d_in[5];
  const float* bv      = (const float*)d_in[6];
  const float* conv1_v = (const float*)d_in[7];
  const float* conv1_g = (const float*)d_in[8];
  const float* conv1_b = (const float*)d_in[9];
  const float* conv2_v = (const float*)d_in[10];
  const float* conv2_g = (const float*)d_in[11];
  const float* conv2_b = (const float*)d_in[12];
  float* out = (float*)d_out;

  // Workspace layout (floats); total ~146 MB
  float* ws = (float*)d_ws;
  float* q   = ws;                         // 16*1024*256
  float* k   = q   + (size_t)B_ * T_ * 256;
  float* v   = k   + (size_t)B_ * T_ * 256;
  float* sc  = v   + (size_t)B_ * T_ * 256;   // 16*1024*1024 (scores -> attn_w)
  float* oa  = sc  + (size_t)B_ * T_ * T_;    // out_attn [B,C,T]
  float* h1  = oa  + (size_t)B_ * C_ * T_;    // conv1 output
  float* rs  = h1  + (size_t)B_ * C_ * T_;    // [B,T]
  float* wx  = rs  + (size_t)B_ * T_;         // [B,T]
  float* wt1 = wx  + (size_t)B_ * T_;         // 3*256*256
  float* wt2 = wt1 + (size_t)KS_ * C_ * C_;

  // K0: weight-norm + transpose conv weights
  wnorm_kernel<<<1, 256, 0, stream>>>(conv1_v, conv1_g, wt1);
  wnorm_kernel<<<1, 256, 0, stream>>>(conv2_v, conv2_g, wt2);

  // K1: q,k,v projections — 16x64 tile per wave32
  qkv_kernel<<<dim3(64 * 4, 3, B_), 32, 0, stream>>>(
      x, Wq, bq, Wk, bk, Wv, bv, q, k, v);

  // K2: causal-masked scaled scores (KEY-major) — 64x16 tile per wave
  scores_kernel<<<dim3(16 * 64, 1, B_), 32, 0, stream>>>(q, k, sc);

  // K3: softmax over query axis (contiguous rows of SC)
  softmax_col_kernel<<<dim3(T_, B_), 256, 0, stream>>>(sc);

  // K4/K5: residual reweighting weights
  attnsum_kernel<<<dim3(T_ / 256, B_), 256, 0, stream>>>(sc, rs);
  wx_kernel<<<B_, 256, 0, stream>>>(rs, wx);

  // K6: out_attn = attn_w^T applied to v, stored [B,C,T] — 64x16 tile/wave
  attnv_kernel<<<dim3(4 * 64, 1, B_), 32, 0, stream>>>(sc, v, oa);

  // K7: conv1 + relu — 64x16 tile per wave
  conv_kernel<<<dim3(4 * 64, 1, B_), 32, 0, stream>>>(
      oa, wt1, conv1_b, h1, nullptr, nullptr, nullptr, 0);

  // K8: conv2 + relu + residual epilogue -> out
  conv_kernel<<<dim3(4 * 64, 1, B_), 32, 0, stream>>>(
      h1, wt2, conv2_b, nullptr, x, wx, out, 1);

  (void)in_sizes; (void)n_in; (void)out_size; (void)ws_size;
}